// HGT_37598143709635
// MI455X (gfx1250) — compile-verified
//
#include <hip/hip_runtime.h>
#include <hip/hip_bf16.h>

typedef float v2f __attribute__((ext_vector_type(2)));
typedef float v8f __attribute__((ext_vector_type(8)));

static constexpr int HID   = 128;
static constexpr int NCHEM = 30000, NGENE = 12000, NPATH = 4000;
static constexpr int NTOT  = NCHEM + NGENE + NPATH;        // 46000
static constexpr int TROWS = 58000;                         // KA/VM scratch rows

// ---------------------------------------------------------------------------
// Build layer-0 input: X = concat(features, embedding/zeros) per node type
// ---------------------------------------------------------------------------
__global__ void hgt_build_x(const float* __restrict__ xc,
                            const float* __restrict__ xg,
                            const float* __restrict__ xp,
                            const float* __restrict__ gemb,
                            const float* __restrict__ pemb,
                            float* __restrict__ X)
{
    int tid = blockIdx.x * blockDim.x + threadIdx.x;
    if (tid >= NTOT * HID) return;
    int n = tid >> 7, f = tid & 127;
    float v;
    if (n < NCHEM) {
        v = (f < 64) ? xc[n * 64 + f] : 0.f;
    } else if (n < NCHEM + NGENE) {
        int m = n - NCHEM;
        v = (f < 64) ? xg[m * 64 + f] : gemb[f - 64];
    } else {
        int m = n - NCHEM - NGENE;
        v = (f < 64) ? xp[m * 64 + f] : pemb[f - 64];
    }
    X[tid] = v;
}

// ---------------------------------------------------------------------------
// Generic fp32 WMMA GEMM:  C[z] = epilogue( A[z] (M x Kdim) @ W[z] (Kdim x N) )
// One wave per 16x64 output tile: A fragment loaded once per K-step and
// reused across 4 V_WMMA_F32_16X16X4_F32 (4x arithmetic intensity on A).
// mode 0: + bias (optional)                      [projections / predictor]
// mode 1: gated skip: relu(g*gelu(v+b) + (1-g)*Xin), g = sigmoid(*gatePtr)
// A frag (ISA 16x4 f32): lanes 0-15 hold K={k,k+1}, lanes 16-31 K={k+2,k+3}.
// ---------------------------------------------------------------------------
__global__ void __launch_bounds__(128)
hgt_wmma_gemm(const float* __restrict__ A, int lda, int aZ,
              const float* __restrict__ W, int ldw, int wZ,
              const float* __restrict__ bias,
              float* __restrict__ C, int ldc, int cZ,
              const float* __restrict__ Xin,
              const float* __restrict__ gatePtr,
              int M, int Kdim, int Ncols, int mode)
{
    const int lane  = threadIdx.x;
    const int tileM = blockIdx.x * 4 + threadIdx.y;
    if (tileM * 16 >= M) return;
    const int z = blockIdx.z;
    A += (size_t)z * aZ;
    W += (size_t)z * wZ;
    C += (size_t)z * cZ;

    const int half    = lane >> 4;    // 0: K pair {k,k+1}, 1: {k+2,k+3}
    const int l15     = lane & 15;
    const int row     = tileM * 16 + l15;
    const int colBase = blockIdx.y * 64;

    int colLd[4];
#pragma unroll
    for (int s = 0; s < 4; ++s) {
        int c = colBase + s * 16 + l15;
        colLd[s] = (c < Ncols) ? c : 0;   // clamp OOB column loads
    }

    const float* __restrict__ arow = A + (size_t)row * lda;
    v8f acc[4];
#pragma unroll
    for (int s = 0; s < 4; ++s) acc[s] = (v8f){0.f,0.f,0.f,0.f,0.f,0.f,0.f,0.f};

#pragma unroll 4
    for (int k = 0; k < Kdim; k += 4) {
        const int ka = k + half * 2;
        v2f af;
        af.x = arow[ka];
        af.y = arow[ka + 1];
        const float* __restrict__ w0 = W + (size_t)ka * ldw;
        const float* __restrict__ w1 = w0 + ldw;
#pragma unroll
        for (int s = 0; s < 4; ++s) {
            v2f bf;
            bf.x = w0[colLd[s]];
            bf.y = w1[colLd[s]];
            acc[s] = __builtin_amdgcn_wmma_f32_16x16x4_f32(false, af, false, bf,
                                                           (short)0, acc[s],
                                                           false, false);
        }
    }

    float gate = 0.f;
    if (mode == 1) gate = 1.f / (1.f + expf(-gatePtr[0]));

#pragma unroll
    for (int s = 0; s < 4; ++s) {
        const int col = colBase + s * 16 + l15;
        const float b = (bias != nullptr && col < Ncols) ? bias[col] : 0.f;
#pragma unroll
        for (int j = 0; j < 8; ++j) {
            const int rr = tileM * 16 + half * 8 + j;  // C layout: vgpr j -> M=j / j+8
            float v = acc[s][j] + b;
            if (mode == 1 && col < Ncols) {
                const float ge = 0.5f * v * (1.f + erff(v * 0.70710678118654752f));
                const float x0 = Xin[(size_t)rr * ldc + col];
                v = gate * ge + (1.f - gate) * x0;
                v = fmaxf(v, 0.f);                     // inter-layer / pre-pred ReLU
            }
            if (col < Ncols) C[(size_t)rr * ldc + col] = v;
        }
    }
}

// ---------------------------------------------------------------------------
// Edge pass 1: logits a[e,h] = dot(KA[src,h,:], Q[dst,h,:]) * p[h]/sqrt(32)
// and segment max via monotonic-uint atomicMax. float4 gathers (16B aligned).
// ---------------------------------------------------------------------------
__global__ void hgt_edge_logits(const float* __restrict__ KA,
                                const float* __restrict__ Q,
                                const int* __restrict__ src,
                                const int* __restrict__ dst,
                                const float* __restrict__ pr,
                                float* __restrict__ alpha,
                                unsigned* __restrict__ amaxU,
                                int E)
{
    int t = blockIdx.x * blockDim.x + threadIdx.x;
    if (t >= E * 4) return;
    int e = t >> 2, h = t & 3;
    int s = src[e], d = dst[e];
    const float4* ka = (const float4*)(KA + (size_t)s * HID + h * 32);
    const float4* q  = (const float4*)(Q  + (size_t)d * HID + h * 32);
    float acc = 0.f;
#pragma unroll
    for (int i = 0; i < 8; ++i) {
        float4 a4 = ka[i], q4 = q[i];
        acc += a4.x * q4.x + a4.y * q4.y + a4.z * q4.z + a4.w * q4.w;
    }
    float a = acc * pr[h] * 0.17677669529663689f;   // 1/sqrt(DH=32)
    alpha[(size_t)e * 4 + h] = a;
    unsigned u = __float_as_uint(a);
    u = (u & 0x80000000u) ? ~u : (u | 0x80000000u); // order-preserving encode
    atomicMax(&amaxU[(size_t)d * 4 + h], u);
}

// ---------------------------------------------------------------------------
// Edge pass 2: ex = exp(a - amax);  den += ex;  agg += ex * VM[src]  (unnorm.)
// ---------------------------------------------------------------------------
__global__ void hgt_edge_accum(const float* __restrict__ VM,
                               const int* __restrict__ src,
                               const int* __restrict__ dst,
                               const float* __restrict__ alpha,
                               const unsigned* __restrict__ amaxU,
                               float* __restrict__ den,
                               float* __restrict__ agg,
                               int E)
{
    int t = blockIdx.x * blockDim.x + threadIdx.x;
    if (t >= E * 4) return;
    int e = t >> 2, h = t & 3;
    int s = src[e], d = dst[e];
    unsigned u = amaxU[(size_t)d * 4 + h];
    u = (u & 0x80000000u) ? (u & 0x7fffffffu) : ~u; // decode
    float amax = __uint_as_float(u);
    float ex = expf(alpha[(size_t)e * 4 + h] - amax);
    atomicAdd(&den[(size_t)d * 4 + h], ex);
    const float4* vm = (const float4*)(VM + (size_t)s * HID + h * 32);
    float* ag = agg + (size_t)d * HID + h * 32;
#pragma unroll
    for (int i = 0; i < 8; ++i) {
        float4 v4 = vm[i];
        atomicAdd(&ag[i * 4 + 0], ex * v4.x);
        atomicAdd(&ag[i * 4 + 1], ex * v4.y);
        atomicAdd(&ag[i * 4 + 2], ex * v4.z);
        atomicAdd(&ag[i * 4 + 3], ex * v4.w);
    }
}

__global__ void hgt_init(unsigned* __restrict__ amaxU,
                         float* __restrict__ den,
                         float* __restrict__ agg)
{
    int tid = blockIdx.x * blockDim.x + threadIdx.x;
    if (tid < NTOT * HID) agg[tid] = 0.f;
    if (tid < NTOT * 4) { amaxU[tid] = 0x007FFFFFu; den[tid] = 0.f; } // enc(-inf)
}

__global__ void hgt_norm(float* __restrict__ agg, const float* __restrict__ den)
{
    int tid = blockIdx.x * blockDim.x + threadIdx.x;
    if (tid >= NTOT * HID) return;
    int n = tid >> 7, h = (tid >> 5) & 3;
    agg[tid] = agg[tid] / fmaxf(den[n * 4 + h], 1e-16f);
}

// ---------------------------------------------------------------------------
extern "C" void kernel_launch(void* const* d_in, const int* in_sizes, int n_in,
                              void* d_out, int out_size, void* d_ws, size_t ws_size,
                              hipStream_t stream)
{
    (void)n_in; (void)out_size; (void)ws_size;
    const float* xc = (const float*)d_in[0];
    const float* xg = (const float*)d_in[1];
    const float* xp = (const float*)d_in[2];
    const int* esrc[4] = { (const int*)d_in[3], (const int*)d_in[5],
                           (const int*)d_in[7], (const int*)d_in[9] };
    const int* edst[4] = { (const int*)d_in[4], (const int*)d_in[6],
                           (const int*)d_in[8], (const int*)d_in[10] };
    int Ecnt[4], eOff[4];
    { int acc = 0;
      for (int r = 0; r < 4; ++r) { Ecnt[r] = in_sizes[3 + 2 * r]; eOff[r] = acc; acc += Ecnt[r]; } }

    // --- param leaf indices (dict insertion order, flattened after input 10) ---
    int pi = 11;
    int Wk[2][3], bk[2][3], Wq[2][3], bq[2][3], Wv[2][3], bv[2][3];
    int Wo[2][3], bo[2][3], sk[2][3];
    int Ar[2][4], Mr[2][4], Pr[2][4];
    for (int l = 0; l < 2; ++l) {
        for (int t = 0; t < 3; ++t) {
            Wk[l][t] = pi++; bk[l][t] = pi++;
            Wq[l][t] = pi++; bq[l][t] = pi++;
            Wv[l][t] = pi++; bv[l][t] = pi++;
            Wo[l][t] = pi++; bo[l][t] = pi++;
            sk[l][t] = pi++;
        }
        for (int r = 0; r < 4; ++r) { Ar[l][r] = pi++; Mr[l][r] = pi++; Pr[l][r] = pi++; }
    }
    const int iGemb = pi++, iPemb = pi++, iPredW = pi++, iPredB = pi++;
    auto P = [&](int i) { return (const float*)d_in[i]; };

    // --- workspace layout (floats); agg aliases K (dead by the time agg is used)
    float* ws   = (float*)d_ws;
    const size_t nX = (size_t)NTOT * HID;
    float*    X     = ws;
    float*    Kb    = X  + nX;
    float*    Qb    = Kb + nX;
    float*    Vb    = Qb + nX;
    float*    T     = Vb + nX;                         // KA then VM, 58000x128
    float*    alpha = T + (size_t)TROWS * HID;
    unsigned* amaxU = (unsigned*)(alpha + (size_t)(eOff[3] + Ecnt[3]) * 4);
    float*    den   = (float*)(amaxU + (size_t)NTOT * 4);
    float*    agg   = Kb;                              // alias

    const int Nt[3]      = { NCHEM, NGENE, NPATH };
    const int off[3]     = { 0, NCHEM, NCHEM + NGENE };
    const int eSrcT[4]   = { 0, 1, 1, 2 };             // cg, gc, gp, pg
    const int eDstT[4]   = { 1, 0, 2, 1 };
    const int tRowOff[4] = { 0, 30000, 42000, 54000 }; // rows in T per edge type

    auto gemm = [&](const float* A, int lda, int aZ,
                    const float* W, int ldw, int wZ,
                    const float* bias,
                    float* C, int ldc, int cZ,
                    const float* Xin, const float* gate,
                    int M, int Kd, int Ncols, int Z, int mode) {
        dim3 blk(32, 4, 1);
        int mTiles = (M + 15) / 16;
        dim3 grd((mTiles + 3) / 4, (Ncols + 63) / 64, Z);
        hgt_wmma_gemm<<<grd, blk, 0, stream>>>(A, lda, aZ, W, ldw, wZ, bias,
                                               C, ldc, cZ, Xin, gate,
                                               M, Kd, Ncols, mode);
    };

    hgt_build_x<<<(NTOT * HID + 255) / 256, 256, 0, stream>>>(
        xc, xg, xp, P(iGemb), P(iPemb), X);

    for (int l = 0; l < 2; ++l) {
        // 1) K/Q/V projections per node type
        for (int t = 0; t < 3; ++t) {
            const float* At = X + (size_t)off[t] * HID;
            gemm(At, 128, 0, P(Wk[l][t]), 128, 0, P(bk[l][t]),
                 Kb + (size_t)off[t] * HID, 128, 0, nullptr, nullptr,
                 Nt[t], 128, 128, 1, 0);
            gemm(At, 128, 0, P(Wq[l][t]), 128, 0, P(bq[l][t]),
                 Qb + (size_t)off[t] * HID, 128, 0, nullptr, nullptr,
                 Nt[t], 128, 128, 1, 0);
            gemm(At, 128, 0, P(Wv[l][t]), 128, 0, P(bv[l][t]),
                 Vb + (size_t)off[t] * HID, 128, 0, nullptr, nullptr,
                 Nt[t], 128, 128, 1, 0);
        }
        // 2) KA_r = K_srcType @ a_r  (per-head 32x32, z = head)
        for (int r = 0; r < 4; ++r) {
            int st = eSrcT[r];
            gemm(Kb + (size_t)off[st] * HID, 128, 32,
                 P(Ar[l][r]), 32, 1024, nullptr,
                 T + (size_t)tRowOff[r] * HID, 128, 32, nullptr, nullptr,
                 Nt[st], 32, 32, 4, 0);
        }
        // 3) init amax/den/agg
        hgt_init<<<(NTOT * HID + 255) / 256, 256, 0, stream>>>(amaxU, den, agg);
        // 4) logits + segment max
        for (int r = 0; r < 4; ++r) {
            int dt = eDstT[r], E = Ecnt[r];
            hgt_edge_logits<<<(E * 4 + 255) / 256, 256, 0, stream>>>(
                T + (size_t)tRowOff[r] * HID,
                Qb + (size_t)off[dt] * HID,
                esrc[r], edst[r], P(Pr[l][r]),
                alpha + (size_t)eOff[r] * 4,
                amaxU + (size_t)off[dt] * 4, E);
        }
        // 5) VM_r = V_srcType @ m_r (overwrites T; pass 1 already consumed KA)
        for (int r = 0; r < 4; ++r) {
            int st = eSrcT[r];
            gemm(Vb + (size_t)off[st] * HID, 128, 32,
                 P(Mr[l][r]), 32, 1024, nullptr,
                 T + (size_t)tRowOff[r] * HID, 128, 32, nullptr, nullptr,
                 Nt[st], 32, 32, 4, 0);
        }
        // 6) exp / den / unnormalized aggregation
        for (int r = 0; r < 4; ++r) {
            int dt = eDstT[r], E = Ecnt[r];
            hgt_edge_accum<<<(E * 4 + 255) / 256, 256, 0, stream>>>(
                T + (size_t)tRowOff[r] * HID, esrc[r], edst[r],
                alpha + (size_t)eOff[r] * 4,
                amaxU + (size_t)off[dt] * 4,
                den + (size_t)off[dt] * 4,
                agg + (size_t)off[dt] * HID, E);
        }
        // 7) normalize by segment sum
        hgt_norm<<<(NTOT * HID + 255) / 256, 256, 0, stream>>>(agg, den);
        // 8) out-proj + exact GELU + gated skip + ReLU, in place into X
        for (int t = 0; t < 3; ++t) {
            float* Ct = X + (size_t)off[t] * HID;
            gemm(agg + (size_t)off[t] * HID, 128, 0, P(Wo[l][t]), 128, 0, P(bo[l][t]),
                 Ct, 128, 0, Ct, P(sk[l][t]), Nt[t], 128, 128, 1, 1);
        }
    }
    // predictor: chem rows (offset 0) @ 128x12 + bias -> d_out
    gemm(X, 128, 0, P(iPredW), 12, 0, P(iPredB),
         (float*)d_out, 12, 0, nullptr, nullptr, NCHEM, 128, 12, 1, 0);
}